// GNN_48266842472623
// MI455X (gfx1250) — compile-verified
//
#include <hip/hip_runtime.h>

// GraphSAGE (4 layers, mean aggregation) for MI455X / gfx1250.
// N=100000 nodes, H=256 features, E=3200000 edges, L=4.
//
// Pipeline per layer (all on `stream`):
//   1) zero agg buffer
//   2) scatter: agg[dst] += h[src]           (f32 atomics, L2-resident: 102MB < 192MB L2)
//   3) WMMA GEMM: out = (agg * 1/deg) @ Wl^T + h @ Wr^T + b  [+ relu]
//      - uses V_WMMA_F32_16X16X4_F32 (true fp32 matrix op)
//      - writes in place over agg (safe: per-block 16-row tile, barrier before store)
// Degrees computed once (dst-only dependence).

#define HDIM 256
#define NNODES 100000
#define NEDGES 3200000
#define NLAYERS 4

typedef __attribute__((ext_vector_type(2))) float v2f;
typedef __attribute__((ext_vector_type(8))) float v8f;

__global__ void zero_kernel(float4* __restrict__ p, int n4) {
    int i = blockIdx.x * blockDim.x + threadIdx.x;
    if (i < n4) p[i] = make_float4(0.f, 0.f, 0.f, 0.f);
}

__global__ void degree_kernel(const int* __restrict__ dst,
                              float* __restrict__ cnt, int nE) {
    int e = blockIdx.x * blockDim.x + threadIdx.x;
    if (e < nE) atomicAdd(&cnt[dst[e]], 1.0f);
}

// 64 threads per edge, 4 floats per thread (float4 gather + 4 f32 atomic adds).
__global__ void scatter_kernel(const float* __restrict__ feat,
                               const int* __restrict__ src,
                               const int* __restrict__ dst,
                               float* __restrict__ agg) {
    long long tid = (long long)blockIdx.x * blockDim.x + threadIdx.x;
    int e = (int)(tid >> 6);
    if (e >= NEDGES) return;
    int c = ((int)tid & 63) * 4;
    int s = src[e];
    int d = dst[e];
    const float4 v = *(const float4*)(feat + (size_t)s * HDIM + c);
    float* p = agg + (size_t)d * HDIM + c;
    atomicAdd(p + 0, v.x);
    atomicAdd(p + 1, v.y);
    atomicAdd(p + 2, v.z);
    atomicAdd(p + 3, v.w);
}

// One wave per 16x16 output tile; 16 waves per block cover one 16-row strip
// across all 16 column tiles (block reads Wl/Wr exactly once; A rows hit WGP$).
// out may alias agg: __syncthreads() separates all A-reads from tile stores,
// and blocks touch disjoint 16-row strips.
__global__ void __launch_bounds__(512)
sage_gemm_kernel(const float* __restrict__ agg,   // neighbor sums [N,H]
                 const float* __restrict__ hfeat, // self features [N,H]
                 const float* __restrict__ cnt,   // in-degree [N]
                 const float* __restrict__ Wl,    // [H,H] row-major (lin_l)
                 const float* __restrict__ Wr,    // [H,H] row-major (lin_r)
                 const float* __restrict__ bias,  // [H]
                 float* __restrict__ out,         // [N,H] (may alias agg)
                 int do_relu) {
    const int wave = threadIdx.x >> 5;   // 0..15 -> column tile
    const int lane = threadIdx.x & 31;
    const int m = lane & 15;             // row (A) / col (B,D) within tile
    const int g = lane >> 4;             // half-wave group
    const int row0 = blockIdx.x * 16;
    const int col0 = wave * 16;

    // mean = agg / max(deg,1): fold scale into the A-fragment load.
    const float inv = 1.0f / fmaxf(cnt[row0 + m], 1.0f);

    // Fragment base pointers. A frag (lane m,g): v[r] = A[m][k + 2g + r].
    // B frag (lane n=m,g): v[r] = W[n][k + 2g + r]  (B[k][n] = W[n*H + k]).
    const float* aRow  = agg   + (size_t)(row0 + m) * HDIM + 2 * g;
    const float* hRow  = hfeat + (size_t)(row0 + m) * HDIM + 2 * g;
    const float* wlRow = Wl    + (size_t)(col0 + m) * HDIM + 2 * g;
    const float* wrRow = Wr    + (size_t)(col0 + m) * HDIM + 2 * g;

    v8f acc = {};
#pragma unroll 4
    for (int k = 0; k < HDIM; k += 4) {
        v2f a = *(const v2f*)(aRow + k);
        a *= inv;
        v2f bl = *(const v2f*)(wlRow + k);
        acc = __builtin_amdgcn_wmma_f32_16x16x4_f32(
            false, a, false, bl, (short)0, acc, false, false);
        v2f h  = *(const v2f*)(hRow + k);
        v2f br = *(const v2f*)(wrRow + k);
        acc = __builtin_amdgcn_wmma_f32_16x16x4_f32(
            false, h, false, br, (short)0, acc, false, false);
    }

    const float bb = bias[col0 + m];

    // Ensure every wave is done reading its agg tile before in-place store.
    __syncthreads();

    // D layout: lane (n=m, g), VGPR r -> row (r + 8g), col n.
    float* orow = out + (size_t)(row0 + 8 * g) * HDIM + col0 + m;
#pragma unroll
    for (int r = 0; r < 8; ++r) {
        float v = acc[r] + bb;
        if (do_relu) v = fmaxf(v, 0.0f);
        orow[(size_t)r * HDIM] = v;
    }
}

extern "C" void kernel_launch(void* const* d_in, const int* in_sizes, int n_in,
                              void* d_out, int out_size, void* d_ws, size_t ws_size,
                              hipStream_t stream) {
    const float* x   = (const float*)d_in[0];            // [N,H]
    const int*   ei  = (const int*)d_in[1];              // [2,E]: src then dst
    const float* Wl  = (const float*)d_in[2];            // [L,H,H]
    const float* Wr  = (const float*)d_in[3];            // [L,H,H]
    const float* bia = (const float*)d_in[4];            // [L,H]
    float* out = (float*)d_out;                          // [N,H]

    const int* src = ei;
    const int* dst = ei + NEDGES;

    // Workspace: one N*H float buffer (ping) + N float degree vector.
    float* bufA = (float*)d_ws;
    float* cnt  = bufA + (size_t)NNODES * HDIM;

    const int featN4 = NNODES * HDIM / 4;                // 6,400,000 float4
    const int cntN4  = NNODES / 4;                       // 25,000 float4

    // Degrees: computed once (dst-only).
    zero_kernel<<<(cntN4 + 255) / 256, 256, 0, stream>>>((float4*)cnt, cntN4);
    degree_kernel<<<(NEDGES + 255) / 256, 256, 0, stream>>>(dst, cnt, NEDGES);

    const long long scatterThreads = (long long)NEDGES * 64;
    const int scatterBlocks = (int)((scatterThreads + 255) / 256);

    const float* hin = x;
    for (int l = 0; l < NLAYERS; ++l) {
        float* agg = (l & 1) ? out : bufA;   // l=3 -> out: final result lands in d_out
        zero_kernel<<<(featN4 + 255) / 256, 256, 0, stream>>>((float4*)agg, featN4);
        scatter_kernel<<<scatterBlocks, 256, 0, stream>>>(hin, src, dst, agg);
        sage_gemm_kernel<<<NNODES / 16, 512, 0, stream>>>(
            agg, hin, cnt,
            Wl + (size_t)l * HDIM * HDIM,
            Wr + (size_t)l * HDIM * HDIM,
            bia + (size_t)l * HDIM,
            agg, (l < NLAYERS - 1) ? 1 : 0);
        hin = agg;
    }
}